// SparseNGCNLayer_48541720379663
// MI455X (gfx1250) — compile-verified
//
#include <hip/hip_runtime.h>
#include <hip/hip_bf16.h>

typedef __attribute__((ext_vector_type(2))) float v2f;
typedef __attribute__((ext_vector_type(8))) float v8f;

#define IN_CH  256
#define OUT_CH 64
#define EPW    8            // edges per wave in the SpMM hop

// ---------------------------------------------------------------------------
// Dense transform: out[r][n] = relu(sum_k feat[r][k]*W[k][n] + bias[n])
// One wave -> 16x64 tile via 4 f32 WMMA accumulators (bias pre-loaded into C),
// K stepped by 4. Weight staged in LDS transposed + XOR-swizzled so B-frag
// reads are bank-conflict-free ds_load_b64.
// ---------------------------------------------------------------------------
__global__ __launch_bounds__(256) void ngcn_gemm_relu(
    const float* __restrict__ feat, const float* __restrict__ weight,
    const float* __restrict__ bias, float* __restrict__ out, int nrows)
{
    __shared__ float wT[OUT_CH * IN_CH];   // exactly 64 KB

    // Cooperative transposed + swizzled store: wT[n][k ^ (4*(n&15))] = W[k][n]
    for (int i = threadIdx.x; i < IN_CH * OUT_CH; i += 256) {
        int k = i >> 6;     // 0..255
        int n = i & 63;     // 0..63
        wT[n * IN_CH + (k ^ ((n & 15) << 2))] = weight[i];
    }
    __syncthreads();

    const int lane = threadIdx.x & 31;
    const int wave = threadIdx.x >> 5;
    const int tile = blockIdx.x * 8 + wave;          // 16-row tile index
    if (tile * 16 >= nrows) return;                  // wave-uniform: EXEC stays all-1s

    const int m    = lane & 15;                      // M index (and N col for B/C/D)
    const int h    = lane >> 4;                      // lane half selects K pair
    const int row0 = tile * 16;
    const int rclamp = (row0 + m < nrows) ? (row0 + m) : (nrows - 1);
    const float* arow = feat + (size_t)rclamp * IN_CH;
    const int swz = m << 2;                          // XOR swizzle for this lane

    // Bias folded into the accumulators: C[M][N=m] column is constant per lane.
    const float b0v = bias[m], b1v = bias[16 + m], b2v = bias[32 + m], b3v = bias[48 + m];
    v8f acc0 = {b0v, b0v, b0v, b0v, b0v, b0v, b0v, b0v};
    v8f acc1 = {b1v, b1v, b1v, b1v, b1v, b1v, b1v, b1v};
    v8f acc2 = {b2v, b2v, b2v, b2v, b2v, b2v, b2v, b2v};
    v8f acc3 = {b3v, b3v, b3v, b3v, b3v, b3v, b3v, b3v};

#pragma unroll 4
    for (int k0 = 0; k0 < IN_CH; k0 += 4) {
        const int ka = k0 + 2 * h;                   // this half's K pair start
        v2f a  = *(const v2f*)(arow + ka);           // A frag: feat[m][ka..ka+1]
        const int kk = ka ^ swz;                     // swizzled LDS column
        v2f b0 = *(const v2f*)(&wT[(m +  0) * IN_CH + kk]);
        v2f b1 = *(const v2f*)(&wT[(m + 16) * IN_CH + kk]);
        v2f b2 = *(const v2f*)(&wT[(m + 32) * IN_CH + kk]);
        v2f b3 = *(const v2f*)(&wT[(m + 48) * IN_CH + kk]);
        acc0 = __builtin_amdgcn_wmma_f32_16x16x4_f32(false, a, false, b0, (short)0, acc0, false, false);
        acc1 = __builtin_amdgcn_wmma_f32_16x16x4_f32(false, a, false, b1, (short)0, acc1, false, false);
        acc2 = __builtin_amdgcn_wmma_f32_16x16x4_f32(false, a, false, b2, (short)0, acc2, false, false);
        acc3 = __builtin_amdgcn_wmma_f32_16x16x4_f32(false, a, false, b3, (short)0, acc3, false, false);
    }

    // Epilogue: ReLU + store. D layout: VGPR r holds row (r + 8*h), col = m.
    v8f accs[4] = {acc0, acc1, acc2, acc3};
    float* obase = out + ((size_t)row0 + 8 * h) * OUT_CH + m;
    if (row0 + 16 <= nrows) {
        // Full tile (always taken for N % 16 == 0): branch-free stores.
#pragma unroll
        for (int nt = 0; nt < 4; ++nt)
#pragma unroll
            for (int r = 0; r < 8; ++r) {
                float val = accs[nt][r];
                obase[(size_t)r * OUT_CH + nt * 16] = val > 0.f ? val : 0.f;
            }
    } else {
        // Partial tail tile: per-row guard.
#pragma unroll
        for (int nt = 0; nt < 4; ++nt)
#pragma unroll
            for (int r = 0; r < 8; ++r) {
                if (row0 + 8 * h + r < nrows) {
                    float val = accs[nt][r];
                    obase[(size_t)r * OUT_CH + nt * 16] = val > 0.f ? val : 0.f;
                }
            }
    }
}

// ---------------------------------------------------------------------------
// One SpMM hop: dst[rows[e]][c] += vals[e] * src[cols[e]][c]
// One wave handles EPW edges; edge metadata is wave-uniform -> scalar loads.
// Lanes cover 2 channels each: 8B coalesced gather + 2 f32 atomics (L2).
// ---------------------------------------------------------------------------
__global__ __launch_bounds__(256) void ngcn_spmm_atomic(
    const long long* __restrict__ rows, const long long* __restrict__ cols,
    const float* __restrict__ vals, const float* __restrict__ src,
    float* __restrict__ dst, int nedges)
{
    const int waveId = __builtin_amdgcn_readfirstlane(
        blockIdx.x * 8 + (threadIdx.x >> 5));        // provably uniform
    const int ch = (threadIdx.x & 31) << 1;          // channel pair 0,2,...,62
    const long long e0 = (long long)waveId * EPW;

    for (int i = 0; i < EPW; ++i) {
        const long long e = e0 + i;
        if (e >= nedges) return;                     // scalar-uniform exit
        const long long r = rows[e];                 // s_load_b64
        const long long c = cols[e];                 // s_load_b64
        const float     v = vals[e];                 // s_load_b32

        v2f x = *(const v2f*)(src + c * OUT_CH + ch);
        float* d = dst + r * OUT_CH + ch;
        unsafeAtomicAdd(d,     v * x.x);             // global_atomic_add_f32
        unsafeAtomicAdd(d + 1, v * x.y);
    }
}

// ---------------------------------------------------------------------------
extern "C" void kernel_launch(void* const* d_in, const int* in_sizes, int n_in,
                              void* d_out, int out_size, void* d_ws, size_t ws_size,
                              hipStream_t stream) {
    const long long* adj  = (const long long*)d_in[0];   // [2, E] int64
    const float* vals     = (const float*)d_in[1];       // [E]
    const float* feat     = (const float*)d_in[2];       // [N, 256]
    const float* weight   = (const float*)d_in[3];       // [256, 64]
    const float* bias     = (const float*)d_in[4];       // [1, 64]
    float* out = (float*)d_out;
    float* tmp = (float*)d_ws;                           // needs N*64*4 = 25.6 MB

    const int nE = in_sizes[1];                          // 1,600,000
    const int nN = in_sizes[2] / IN_CH;                  // 100,000
    const long long* rows = adj;
    const long long* cols = adj + nE;

    const size_t feat_bytes = (size_t)nN * OUT_CH * sizeof(float);

    // 1) base = relu(feat @ W + b)  -> tmp
    const int tiles   = (nN + 15) / 16;
    const int gblocks = (tiles + 7) / 8;
    ngcn_gemm_relu<<<gblocks, 256, 0, stream>>>(feat, weight, bias, tmp, nN);

    // waves = ceil(E / EPW); 8 waves per block
    const int waves   = (nE + EPW - 1) / EPW;
    const int sblocks = (waves + 7) / 8;

    // 2) hop 1: tmp -> out
    hipMemsetAsync(out, 0, feat_bytes, stream);
    ngcn_spmm_atomic<<<sblocks, 256, 0, stream>>>(rows, cols, vals, tmp, out, nE);

    // 3) hop 2: out -> tmp
    hipMemsetAsync(tmp, 0, feat_bytes, stream);
    ngcn_spmm_atomic<<<sblocks, 256, 0, stream>>>(rows, cols, vals, out, tmp, nE);

    // 4) hop 3: tmp -> out  (final result)
    hipMemsetAsync(out, 0, feat_bytes, stream);
    ngcn_spmm_atomic<<<sblocks, 256, 0, stream>>>(rows, cols, vals, tmp, out, nE);
}